// DegradationModel_53171695124493
// MI455X (gfx1250) — compile-verified
//
#include <hip/hip_runtime.h>
#include <cstddef>

typedef __attribute__((ext_vector_type(2))) float v2f;
typedef __attribute__((ext_vector_type(8))) float v8f;

// Problem sizes: B=256, MU=64, NX=8, W=64, DEPTH=3 ; N = 16384 rows.

// ---- output offsets (floats, concat order of reference tuple) ----
static constexpr size_t OFF_RES    = 0;          // 256*64*8  = 131072
static constexpr size_t OFF_MEAN   = 131072;     // 256*64
static constexpr size_t OFF_LSIG   = 147456;     // 256*64
static constexpr size_t OFF_COEF   = 163840;     // 16384
static constexpr size_t OFF_DCYC   = 180224;     // 16384
static constexpr size_t OFF_D2CYC  = 196608;     // 16384 (zeros)
static constexpr size_t OFF_DOTH   = 212992;     // 16384*3
static constexpr size_t OFF_D2OTH  = 262144;     // 16384*9 (zeros)
static constexpr size_t OFF_DFEAT  = 409600;     // 16384*64
static constexpr size_t OFF_D2FEAT = 1458176;    // 16384*4096 (zeros except [.,0,0])
static constexpr size_t N_D2FEAT   = 67108864;

// ---- LDS layout (float indices); activation row stride 68 avoids bank conflicts ----
static constexpr int ST    = 68;
static constexpr int O_X   = 0;       // 64 x 68 (input, cols 0..66 used, 67 = 0 pad)
static constexpr int O_H0  = 4352;    // h0 / later u0
static constexpr int O_H1  = 8704;    // h1 / later u1
static constexpr int O_H2  = 13056;   // h2 / later u2
static constexpr int O_H3  = 17408;   // h3 / later u3
static constexpr int O_W0  = 21760;   // 80 x 68 (W0 row-major, rows 67..79 zero)
static constexpr int O_WS0 = 27200;   // 64 x 68 row-major
static constexpr int O_WS1 = 31552;
static constexpr int O_WS2 = 35904;
static constexpr int O_B0  = 40256;   // 64
static constexpr int O_BS  = 40320;   // 3*64
static constexpr int O_WF  = 40512;   // 64
static constexpr int O_FT  = 40576;   // 64 (features, shared per block)
static constexpr int O_VAL = 40640;   // 64 (val per row)
static constexpr int O_GX  = 40704;   // 64 x 4 (g_x cols 0..3 per row)
static constexpr int SMEM_FLOATS = 40960;
static constexpr int SMEM_BYTES  = SMEM_FLOATS * 4;  // 160 KB dynamic LDS

__device__ __forceinline__ v8f wmma4(v2f a, v2f b, v8f c) {
  // D(16x16,f32) = A(16x4,f32) * B(4x16,f32) + C  -- exact f32 matrix op
  return __builtin_amdgcn_wmma_f32_16x16x4_f32(false, a, false, b, (short)0, c,
                                               false, false);
}

// Forward layer: Hout(16x64) = relu(Hin(16xK) @ Wr(Kx64 row-major, stride ST) + bias)
// Hin/Hout already offset to this wave's base row (stride ST).
__device__ __forceinline__ void fwd_layer(const float* __restrict__ Hin,
                                          float* __restrict__ Hout,
                                          const float* __restrict__ Wr,
                                          const float* __restrict__ bias,
                                          int K, int sub, int hi) {
  v8f acc[4] = {};
  for (int k = 0; k < K; k += 4) {
    const int kr = k + 2 * hi;                        // A/B k-pair per half-wave
    v2f a = *(const v2f*)(Hin + sub * ST + kr);       // A[m=sub][kr..kr+1]
    const float* w0 = Wr + kr * ST;
    const float* w1 = Wr + (kr + 1) * ST;
#pragma unroll
    for (int t = 0; t < 4; ++t) {
      v2f bb;
      bb.x = w0[t * 16 + sub];                        // B[kr][n]
      bb.y = w1[t * 16 + sub];                        // B[kr+1][n]
      acc[t] = wmma4(a, bb, acc[t]);
    }
  }
#pragma unroll
  for (int t = 0; t < 4; ++t) {
    const float bv = bias[t * 16 + sub];
#pragma unroll
    for (int v = 0; v < 8; ++v) {
      const float z = acc[t][v] + bv;                 // D[M=v+8*hi][N=t*16+sub]
      Hout[(v + 8 * hi) * ST + t * 16 + sub] = fmaxf(z, 0.f);
    }
  }
}

// Backward layer: Hm <- (Uin(16x64) @ Wr^T) * (Hm > 0)   (Wr 64x64 row-major)
__device__ __forceinline__ void bwd_layer(const float* __restrict__ Uin,
                                          float* __restrict__ Hm,
                                          const float* __restrict__ Wr,
                                          int sub, int hi) {
  v8f acc[4] = {};
  for (int k = 0; k < 64; k += 4) {
    const int kc = k + 2 * hi;
    v2f a = *(const v2f*)(Uin + sub * ST + kc);
#pragma unroll
    for (int t = 0; t < 4; ++t) {
      // B[kc][n] = Wr[n][kc]  -> contiguous pair in row-major Wr
      v2f bb = *(const v2f*)(Wr + (t * 16 + sub) * ST + kc);
      acc[t] = wmma4(a, bb, acc[t]);
    }
  }
#pragma unroll
  for (int t = 0; t < 4; ++t) {
#pragma unroll
    for (int v = 0; v < 8; ++v) {
      const int off = (v + 8 * hi) * ST + t * 16 + sub;
      const float hm = Hm[off];
      Hm[off] = (hm > 0.f) ? acc[t][v] : 0.f;
    }
  }
}

__global__ void __launch_bounds__(128)
degmodel_kernel(const float* __restrict__ centers,   // (256,3)
                const int*   __restrict__ indecies,  // (256,)
                const float* __restrict__ xs,        // (256,8)
                const float* __restrict__ mu,        // (64,)
                const float* __restrict__ eps,       // (1000,64)
                const float* __restrict__ dict_kernel, // (1000,128)
                const float* __restrict__ W0g,       // (67,64)
                const float* __restrict__ b0g,       // (64,)
                const float* __restrict__ Wsg,       // (3,64,64)
                const float* __restrict__ bsg,       // (3,64)
                const float* __restrict__ Wfg,       // (64,1)
                const float* __restrict__ bfg,       // (1,)
                float* __restrict__ out) {
  extern __shared__ float smem[];
  const int tid = threadIdx.x;
  const int b = blockIdx.x;          // one sample per block; rows m = 0..63 (mu)
  const int lane = tid & 31;
  const int wv = tid >> 5;           // 4 waves, wave32
  const int sub = lane & 15;
  const int hi = lane >> 4;
  const int R = wv * 16;             // this wave's row base

  float* X   = smem + O_X;
  float* H0  = smem + O_H0;
  float* H1  = smem + O_H1;
  float* H2  = smem + O_H2;
  float* H3  = smem + O_H3;
  float* W0R = smem + O_W0;
  float* WS0 = smem + O_WS0;
  float* WS1 = smem + O_WS1;
  float* WS2 = smem + O_WS2;
  float* B0L = smem + O_B0;
  float* BSL = smem + O_BS;
  float* WFL = smem + O_WF;
  float* FTL = smem + O_FT;
  float* VALL = smem + O_VAL;
  float* GXL = smem + O_GX;

  // ---- Phase A1: features + stage weights in LDS ----
  const int idx = indecies[b];
  if (tid < 64) {
    const float mn = dict_kernel[idx * 128 + tid];
    const float ls = dict_kernel[idx * 128 + 64 + tid];
    out[OFF_MEAN + (size_t)b * 64 + tid] = mn;
    out[OFF_LSIG + (size_t)b * 64 + tid] = ls;
    FTL[tid] = fmaf(expf(ls * 0.5f), eps[idx * 64 + tid], mn);
    B0L[tid] = b0g[tid];
    WFL[tid] = Wfg[tid];
  }
  if (tid < 64 + 64) { /* keep all lanes busy below */ }
  for (int i = tid; i < 80 * 64; i += 128) {   // W0 row-major, zero-padded rows
    const int r = i >> 6, c2 = i & 63;
    W0R[r * ST + c2] = (r < 67) ? W0g[r * 64 + c2] : 0.f;
  }
  for (int i = tid; i < 3 * 64 * 64; i += 128) {
    const int l = i >> 12, rem = i & 4095, r = rem >> 6, c2 = rem & 63;
    smem[O_WS0 + l * 4352 + r * ST + c2] = Wsg[i];
  }
  if (tid < 192) BSL[tid] = bsg[tid];
  __syncthreads();

  const float c_s = centers[b * 3 + 0];
  const float o0  = centers[b * 3 + 1];
  const float o1  = centers[b * 3 + 2];
  const float e_s = expf(-FTL[0]);
  const float a_s = 1e-10f + e_s;

  // ---- Phase A2: build X (64 x 68) ----
  for (int i = tid; i < 64 * ST; i += 128) {
    const int m = i / ST, col = i - m * ST;
    float v;
    if (col == 0)      v = c_s * a_s;
    else if (col == 1) v = o0;
    else if (col == 2) v = o1;
    else if (col == 3) v = mu[m];
    else if (col < 67) v = FTL[col - 3];   // ft[1..63]
    else               v = 0.f;            // K pad
    X[i] = v;
  }
  __syncthreads();

  // ---- Forward MLP (WMMA f32 16x16x4) ----
  fwd_layer(X  + R * ST, H0 + R * ST, W0R, B0L,       68, sub, hi);
  __syncthreads();
  fwd_layer(H0 + R * ST, H1 + R * ST, WS0, BSL,       64, sub, hi);
  __syncthreads();
  fwd_layer(H1 + R * ST, H2 + R * ST, WS1, BSL + 64,  64, sub, hi);
  __syncthreads();
  fwd_layer(H2 + R * ST, H3 + R * ST, WS2, BSL + 128, 64, sub, hi);
  __syncthreads();

  // ---- val = h3.Wf + bf ; H3 <- u3 = Wf * (h3 > 0)  (in place) ----
  if (tid < 64) {
    float* h3 = H3 + tid * ST;
    float acc = bfg[0];
    for (int j = 0; j < 64; ++j) {
      const float t = h3[j];
      acc = fmaf(t, WFL[j], acc);
      h3[j] = (t > 0.f) ? WFL[j] : 0.f;
    }
    VALL[tid] = acc;
  }
  __syncthreads();

  // ---- Backward: u2, u1, u0 (masked GEMMs, in place over H2/H1/H0) ----
  bwd_layer(H3 + R * ST, H2 + R * ST, WS2, sub, hi);
  __syncthreads();
  bwd_layer(H2 + R * ST, H1 + R * ST, WS1, sub, hi);
  __syncthreads();
  bwd_layer(H1 + R * ST, H0 + R * ST, WS0, sub, hi);
  __syncthreads();

  // ---- g_x (16 x 67) = U0 @ W0^T : 5 N-tiles; stream results straight out ----
  {
    v8f acc[5] = {};
    const float* U0 = H0 + R * ST;
    for (int k = 0; k < 64; k += 4) {
      const int kc = k + 2 * hi;
      v2f a = *(const v2f*)(U0 + sub * ST + kc);
#pragma unroll
      for (int t = 0; t < 5; ++t) {
        v2f bb = *(const v2f*)(W0R + (t * 16 + sub) * ST + kc);  // B[kc][n]=W0[n][kc]
        acc[t] = wmma4(a, bb, acc[t]);
      }
    }
#pragma unroll
    for (int t = 0; t < 5; ++t) {
      const int n = t * 16 + sub;
      if (n < 4) {                       // g_x[0..3] -> LDS for scalar outputs
#pragma unroll
        for (int v = 0; v < 8; ++v)
          GXL[(R + v + 8 * hi) * 4 + n] = acc[t][v];
      } else if (n < 67) {               // dresdfeatures[., n-3], cols 1..63
#pragma unroll
        for (int v = 0; v < 8; ++v) {
          const int m = R + v + 8 * hi;
          out[OFF_DFEAT + (size_t)(b * 64 + m) * 64 + (n - 3)] = acc[t][v];
        }
      }
    }
  }
  __syncthreads();

  // ---- Per-row scalar outputs (Hessian blocks are zero except Hf[0,0]) ----
  if (tid < 64) {
    const int m = tid;
    const int n = b * 64 + m;
    const float g0 = GXL[m * 4 + 0];
    const float g1 = GXL[m * 4 + 1];
    const float g2 = GXL[m * 4 + 2];
    const float g3 = GXL[m * 4 + 3];
    const float val = VALL[m];
    const float gc = g0 * a_s;
    const float ce = c_s * e_s;
    out[OFF_COEF + n] = val;
    out[OFF_DCYC + n] = gc;
    out[OFF_D2CYC + n] = 0.f;                       // Hc == 0 exactly
    out[OFF_DOTH + (size_t)n * 3 + 0] = g1;
    out[OFF_DOTH + (size_t)n * 3 + 1] = g2;
    out[OFF_DOTH + (size_t)n * 3 + 2] = g3;
#pragma unroll
    for (int q = 0; q < 9; ++q)
      out[OFF_D2OTH + (size_t)n * 9 + q] = 0.f;     // Ho == 0 exactly
    out[OFF_DFEAT + (size_t)n * 64 + 0] = -g0 * ce; // gf[0]
    out[OFF_D2FEAT + (size_t)n * 4096] = g0 * ce;   // Hf[0,0]; rest zero-filled
#pragma unroll
    for (int x = 0; x < 8; ++x)                     // res = val + gc*(xs - c)
      out[OFF_RES + (size_t)b * 512 + m * 8 + x] =
          fmaf(gc, xs[b * 8 + x] - c_s, val);
  }
}

// Wide zero-fill for the 256 MB d2resdfeatures block (bandwidth-dominant part).
__global__ void __launch_bounds__(256)
zero_f4_kernel(float4* __restrict__ dst, size_t n4) {
  size_t i = (size_t)blockIdx.x * blockDim.x + threadIdx.x;
  const size_t stride = (size_t)gridDim.x * blockDim.x;
  const float4 z = make_float4(0.f, 0.f, 0.f, 0.f);
  for (; i < n4; i += stride) dst[i] = z;
}

extern "C" void kernel_launch(void* const* d_in, const int* in_sizes, int n_in,
                              void* d_out, int out_size, void* d_ws, size_t ws_size,
                              hipStream_t stream) {
  const float* centers = (const float*)d_in[0];
  const int*   indecies = (const int*)d_in[1];
  const float* xs   = (const float*)d_in[2];
  const float* mu   = (const float*)d_in[3];
  const float* eps  = (const float*)d_in[4];
  const float* dict = (const float*)d_in[5];
  const float* W0   = (const float*)d_in[6];
  const float* b0   = (const float*)d_in[7];
  const float* Ws   = (const float*)d_in[8];
  const float* bs   = (const float*)d_in[9];
  const float* Wf   = (const float*)d_in[10];
  const float* bf   = (const float*)d_in[11];
  float* out = (float*)d_out;

  (void)hipFuncSetAttribute(reinterpret_cast<const void*>(&degmodel_kernel),
                            hipFuncAttributeMaxDynamicSharedMemorySize, SMEM_BYTES);

  // 1) zero-fill d2resdfeatures (16M float4), then 2) main kernel overwrites [.,0,0]
  zero_f4_kernel<<<8192, 256, 0, stream>>>((float4*)(out + OFF_D2FEAT), N_D2FEAT / 4);
  degmodel_kernel<<<256, 128, SMEM_BYTES, stream>>>(
      centers, indecies, xs, mu, eps, dict, W0, b0, Ws, bs, Wf, bf, out);
}